// MultiPerspectiveLayer_53584011985089
// MI455X (gfx1250) — compile-verified
//
#include <hip/hip_runtime.h>
#include <math.h>

// ---------------- problem constants (from reference setup_inputs) ----------
#define NB 32      // batch
#define LL 128     // left length
#define LR 128     // right length
#define DD 512     // feature dim
#define MM 64      // perspectives
#define AA 128     // attention dim
#define CO 259     // output channels: 65 + 64 + 65 + 65

typedef __attribute__((ext_vector_type(2))) float v2f;
typedef __attribute__((ext_vector_type(8))) float v8f;

// V_WMMA_F32_16X16X4_F32: D(16x16,f32) = A(16x4,f32) x B(4x16,f32) + C
// A/B = 2 VGPRs/lane (float2), C/D = 8 VGPRs/lane.
// Lane layout (ISA 7.12.2): half = lane>>4, row/col = lane&15, kk = 2*half.
//   A: a.x = A[row][kk],   a.y = A[row][kk+1]
//   B: b.x = B[kk][col],   b.y = B[kk+1][col]
//   D: acc[j] -> element [j + 8*half][col]
__device__ __forceinline__ v8f wmma4(v2f a, v2f b, v8f c) {
  return __builtin_amdgcn_wmma_f32_16x16x4_f32(false, a, false, b,
                                               (short)0, c, false, false);
}

// gfx1250 has a hardware V_TANH_F32 transcendental; use it if the builtin
// exists, otherwise fall back to libm (safe either way).
__device__ __forceinline__ float fast_tanh(float x) {
#if __has_builtin(__builtin_amdgcn_tanhf)
  return __builtin_amdgcn_tanhf(x);
#elif __has_builtin(__builtin_amdgcn_tanh_f32)
  return __builtin_amdgcn_tanh_f32(x);
#else
  return tanhf(x);
#endif
}

// ---------------------------------------------------------------------------
// h_rt = concat(h_rt_fw, h_rt_bw)  -> (B, D)
__global__ void hcat_kernel(const float* __restrict__ fw,
                            const float* __restrict__ bw,
                            float* __restrict__ h) {
  int i = blockIdx.x * blockDim.x + threadIdx.x;
  if (i < NB * DD) {
    int b = i / DD, d = i - b * DD;
    h[i] = (d < DD / 2) ? fw[b * (DD / 2) + d] : bw[b * (DD / 2) + d - DD / 2];
  }
}

// ---------------------------------------------------------------------------
// a_lt = tanh(reps_lt @ W1) * diag ; a_rt = tanh(reps_rt @ W1)
// grid: (LL/16, B, 2) ; block: 256 (8 waves = 8 column tiles of A=128)
__global__ void attn_proj_kernel(const float* __restrict__ reps_lt,
                                 const float* __restrict__ reps_rt,
                                 const float* __restrict__ w1,   // (D, A)
                                 const float* __restrict__ diag, // (A)
                                 float* __restrict__ a_lt,
                                 float* __restrict__ a_rt) {
  const int side = blockIdx.z;                 // 0 = lt, 1 = rt
  const int b = blockIdx.y;
  const int l0 = blockIdx.x * 16;
  const float* reps = side ? reps_rt : reps_lt;
  float* outp = side ? a_rt : a_lt;

  const int wave = threadIdx.x >> 5, lane = threadIdx.x & 31;
  const int half = lane >> 4, row = lane & 15, kk = half * 2, col = lane & 15;
  const int n0 = wave * 16;

  const float* arow = reps + ((size_t)b * LL + l0 + row) * DD;
  v8f acc = {};
  for (int d = 0; d < DD; d += 4) {
    v2f a;  a.x = arow[d + kk];                    a.y = arow[d + kk + 1];
    v2f bb; bb.x = w1[(size_t)(d + kk) * AA + n0 + col];
            bb.y = w1[(size_t)(d + kk + 1) * AA + n0 + col];
    acc = wmma4(a, bb, acc);
  }
  const float dw = diag[n0 + col];
#pragma unroll
  for (int j = 0; j < 8; ++j) {
    float v = fast_tanh(acc[j]);
    if (side == 0) v *= dw;
    outp[((size_t)b * LL + l0 + j + half * 8) * AA + n0 + col] = v;
  }
}

// ---------------------------------------------------------------------------
// scores = a_lt @ a_rt^T ; probs = softmax_r(scores) ; att = probs @ reps_rt
// grid: (LL/16, B) ; block: 256 (8 waves)
__global__ void attention_kernel(const float* __restrict__ a_lt,
                                 const float* __restrict__ a_rt,
                                 const float* __restrict__ reps_rt,
                                 float* __restrict__ att) {
  __shared__ float sc[16][LR];                 // 8 KB score/prob tile
  const int b = blockIdx.y;
  const int l0 = blockIdx.x * 16;
  const int wave = threadIdx.x >> 5, lane = threadIdx.x & 31;
  const int half = lane >> 4, row = lane & 15, kk = half * 2, col = lane & 15;

  // phase 1: 16(l) x 128(r) raw scores, K-loop over A=128
  {
    const int r0 = wave * 16;
    const float* al = a_lt + ((size_t)b * LL + l0 + row) * AA;
    const float* ar = a_rt + ((size_t)b * LR + r0 + col) * AA;
    v8f acc = {};
    for (int k = 0; k < AA; k += 4) {
      v2f a;  a.x = al[k + kk];  a.y = al[k + kk + 1];
      v2f bb; bb.x = ar[k + kk]; bb.y = ar[k + kk + 1];
      acc = wmma4(a, bb, acc);
    }
#pragma unroll
    for (int j = 0; j < 8; ++j) sc[j + half * 8][r0 + col] = acc[j];
  }
  __syncthreads();

  // phase 2: softmax over r, one row per thread 0..15
  if (threadIdx.x < 16) {
    float* rp = sc[threadIdx.x];
    float mx = -3.4e38f;
    for (int r = 0; r < LR; ++r) mx = fmaxf(mx, rp[r]);
    float s = 0.f;
    for (int r = 0; r < LR; ++r) { float e = expf(rp[r] - mx); rp[r] = e; s += e; }
    const float inv = 1.f / s;
    for (int r = 0; r < LR; ++r) rp[r] *= inv;
  }
  __syncthreads();

  // phase 3: att = probs(16x128) @ reps_rt(128x512); each wave does 4 d-tiles
  for (int i = 0; i < 4; ++i) {
    const int n0 = wave * 16 + i * 128;
    v8f acc = {};
    for (int k = 0; k < LR; k += 4) {
      v2f a;  a.x = sc[row][k + kk]; a.y = sc[row][k + kk + 1];
      v2f bb; bb.x = reps_rt[((size_t)b * LR + k + kk) * DD + n0 + col];
              bb.y = reps_rt[((size_t)b * LR + k + kk + 1) * DD + n0 + col];
      acc = wmma4(a, bb, acc);
    }
#pragma unroll
    for (int j = 0; j < 8; ++j)
      att[((size_t)b * LL + l0 + j + half * 8) * DD + n0 + col] = acc[j];
  }
}

// ---------------------------------------------------------------------------
// cosine-similarity argmax over r (|L| is a positive per-l scale: irrelevant
// for argmax) + gather. grid: (LL, B) ; block: 128 (thread = r)
__global__ void maxatt_kernel(const float* __restrict__ reps_lt,
                              const float* __restrict__ reps_rt,
                              float* __restrict__ max_att) {
  __shared__ float vals[LR];
  __shared__ int idxs[LR];
  const int b = blockIdx.y, l = blockIdx.x, r = threadIdx.x;
  const float4* L = (const float4*)(reps_lt + ((size_t)b * LL + l) * DD);
  const float4* R = (const float4*)(reps_rt + ((size_t)b * LR + r) * DD);
  float dot = 0.f, nr = 0.f;
  for (int d = 0; d < DD / 4; ++d) {
    float4 lv = L[d], rv = R[d];
    dot += lv.x * rv.x + lv.y * rv.y + lv.z * rv.z + lv.w * rv.w;
    nr  += rv.x * rv.x + rv.y * rv.y + rv.z * rv.z + rv.w * rv.w;
  }
  vals[r] = dot / sqrtf(fmaxf(nr, 1e-6f));
  idxs[r] = r;
  __syncthreads();
  for (int s = LR / 2; s > 0; s >>= 1) {
    if (r < s) {
      float v2 = vals[r + s]; int i2 = idxs[r + s];
      if (v2 > vals[r] || (v2 == vals[r] && i2 < idxs[r])) { vals[r] = v2; idxs[r] = i2; }
    }
    __syncthreads();
  }
  const float4* Rb = (const float4*)(reps_rt + ((size_t)b * LR + idxs[0]) * DD);
  float4* Ob = (float4*)(max_att + ((size_t)b * LL + l) * DD);
  for (int d = threadIdx.x; d < DD / 4; d += blockDim.x) Ob[d] = Rb[d];
}

// ---------------------------------------------------------------------------
// _mp_match: prod = L (.) v ; out[off] = tanh(sum_d prod) ;
//            out[off+1+m] = tanh(prod @ K[m,:])
// v_lstride = 0 -> v broadcast over l (shape (B,D)); else (B,LL,D)
// grid: (LL/16, B) ; block: 128 (4 waves = 4 m-tiles of M=64)
__global__ void match_kernel(const float* __restrict__ reps_lt,
                             const float* __restrict__ v, int v_lstride,
                             const float* __restrict__ kern, // (M, D)
                             float* __restrict__ out, int out_off) {
  __shared__ float cpart[16][8];
  const int b = blockIdx.y;
  const int l0 = blockIdx.x * 16;
  const int wave = threadIdx.x >> 5, lane = threadIdx.x & 31;
  const int half = lane >> 4, row = lane & 15, kk = half * 2, col = lane & 15;
  const int m0 = wave * 16;
  const size_t vbase = (size_t)b * (v_lstride ? (size_t)LL * DD : (size_t)DD);

  // cosine partials: thread t handles row (t>>3), d-chunk (t&7) of 64
  {
    const int lr = threadIdx.x >> 3, ch = threadIdx.x & 7;
    const float4* Lr = (const float4*)(reps_lt + ((size_t)b * LL + l0 + lr) * DD + ch * 64);
    const float4* Vr = (const float4*)(v + vbase + (size_t)(l0 + lr) * v_lstride + ch * 64);
    float s = 0.f;
#pragma unroll
    for (int d = 0; d < 16; ++d) {
      float4 lv = Lr[d], vv = Vr[d];
      s += lv.x * vv.x + lv.y * vv.y + lv.z * vv.z + lv.w * vv.w;
    }
    cpart[lr][ch] = s;
  }

  const float* Lrow = reps_lt + ((size_t)b * LL + l0 + row) * DD;
  const float* Vrow = v + vbase + (size_t)(l0 + row) * v_lstride;
  const float* Krow = kern + (size_t)(m0 + col) * DD;

  v8f acc = {};
  for (int d = 0; d < DD; d += 4) {
    v2f a;  a.x = Lrow[d + kk] * Vrow[d + kk];
            a.y = Lrow[d + kk + 1] * Vrow[d + kk + 1];
    v2f bb; bb.x = Krow[d + kk]; bb.y = Krow[d + kk + 1];
    acc = wmma4(a, bb, acc);
  }
#pragma unroll
  for (int j = 0; j < 8; ++j)
    out[((size_t)b * LL + l0 + j + half * 8) * CO + out_off + 1 + m0 + col] =
        fast_tanh(acc[j]);

  __syncthreads();
  if (threadIdx.x < 16) {
    float s = 0.f;
#pragma unroll
    for (int c = 0; c < 8; ++c) s += cpart[threadIdx.x][c];
    out[((size_t)b * LL + l0 + threadIdx.x) * CO + out_off] = fast_tanh(s);
  }
}

// ---------------------------------------------------------------------------
// mpool partial: raw scores s[b,r,l,m] = sum_d L*K*R, max over an r-quarter.
// grid: (LL/16, B, 4) ; block 256 = 8 waves: (m_tile 0..3, r_subhalf 0..1).
// L-tile in LDS; K and L fragments are r-invariant, so r is processed in
// groups of 4 with 4 accumulators -> 4 WMMAs per K/L fragment load.
// tanh is monotone: max raw scores here, tanh once in the combine kernel.
__global__ void mpool_part_kernel(const float* __restrict__ reps_lt,
                                  const float* __restrict__ reps_rt,
                                  const float* __restrict__ kern, // (M, D)
                                  float* __restrict__ mp_part) {  // (4,B,LL,MM)
  __shared__ float Llds[16 * DD];       // 32 KB
  __shared__ float red[2 * 16 * MM];    // 8 KB
  const int b = blockIdx.y;
  const int l0 = blockIdx.x * 16;
  const int q = blockIdx.z;             // r-quarter (32 r's)

  for (int i = threadIdx.x; i < 16 * DD; i += blockDim.x)
    Llds[i] = reps_lt[((size_t)b * LL + l0) * DD + i];
  __syncthreads();

  const int wave = threadIdx.x >> 5, lane = threadIdx.x & 31;
  const int half = lane >> 4, row = lane & 15, kk = half * 2, col = lane & 15;
  const int m0 = (wave & 3) * 16;
  const int sh = wave >> 2;             // sub-half: 16 r's
  const int rbase = q * 32 + sh * 16;

  const float* Lp = Llds + row * DD;
  const float* Kp = kern + (size_t)(m0 + col) * DD;

  float mx[8];
#pragma unroll
  for (int j = 0; j < 8; ++j) mx[j] = -3.4e38f;

  for (int rg = 0; rg < 16; rg += 4) {
    const float* R0 = reps_rt + ((size_t)b * LR + rbase + rg) * DD;
    v8f acc0 = {}, acc1 = {}, acc2 = {}, acc3 = {};
    for (int d = 0; d < DD; d += 4) {
      v2f lf; lf.x = Lp[d + kk]; lf.y = Lp[d + kk + 1];
      v2f bb; bb.x = Kp[d + kk]; bb.y = Kp[d + kk + 1];
      const float4 r0 = *(const float4*)(R0 + d);
      const float4 r1 = *(const float4*)(R0 + DD + d);
      const float4 r2 = *(const float4*)(R0 + 2 * DD + d);
      const float4 r3 = *(const float4*)(R0 + 3 * DD + d);
      v2f a;
      a.x = lf.x * (half ? r0.z : r0.x); a.y = lf.y * (half ? r0.w : r0.y);
      acc0 = wmma4(a, bb, acc0);
      a.x = lf.x * (half ? r1.z : r1.x); a.y = lf.y * (half ? r1.w : r1.y);
      acc1 = wmma4(a, bb, acc1);
      a.x = lf.x * (half ? r2.z : r2.x); a.y = lf.y * (half ? r2.w : r2.y);
      acc2 = wmma4(a, bb, acc2);
      a.x = lf.x * (half ? r3.z : r3.x); a.y = lf.y * (half ? r3.w : r3.y);
      acc3 = wmma4(a, bb, acc3);
    }
#pragma unroll
    for (int j = 0; j < 8; ++j)
      mx[j] = fmaxf(mx[j], fmaxf(fmaxf(acc0[j], acc1[j]), fmaxf(acc2[j], acc3[j])));
  }

#pragma unroll
  for (int j = 0; j < 8; ++j)
    red[sh * 16 * MM + (j + half * 8) * MM + m0 + col] = mx[j];
  __syncthreads();

  for (int i = threadIdx.x; i < 16 * MM; i += blockDim.x) {
    int l = i >> 6, m = i & (MM - 1);
    mp_part[(((size_t)q * NB + b) * LL + l0 + l) * MM + m] =
        fmaxf(red[i], red[16 * MM + i]);
  }
}

// combine 4 quarter-maxes + tanh -> out channels [65,129)
__global__ void mpool_combine_kernel(const float* __restrict__ mp_part,
                                     float* __restrict__ out) {
  int i = blockIdx.x * blockDim.x + threadIdx.x;
  const int n = NB * LL * MM;
  if (i < n) {
    float v = fmaxf(fmaxf(mp_part[i], mp_part[n + i]),
                    fmaxf(mp_part[2 * (size_t)n + i], mp_part[3 * (size_t)n + i]));
    int b = i / (LL * MM), rem = i - b * (LL * MM);
    int l = rem / MM, m = rem - l * MM;
    out[((size_t)b * LL + l) * CO + 65 + m] = fast_tanh(v);
  }
}

// ---------------------------------------------------------------------------
extern "C" void kernel_launch(void* const* d_in, const int* in_sizes, int n_in,
                              void* d_out, int out_size, void* d_ws, size_t ws_size,
                              hipStream_t stream) {
  (void)in_sizes; (void)n_in; (void)out_size; (void)ws_size;
  const float* reps_lt      = (const float*)d_in[0];
  const float* reps_rt      = (const float*)d_in[3];
  const float* h_rt_fw      = (const float*)d_in[4];
  const float* h_rt_bw      = (const float*)d_in[5];
  const float* kernel_full  = (const float*)d_in[6];
  const float* kernel_mpool = (const float*)d_in[7];
  const float* attn_w1      = (const float*)d_in[8];
  const float* diag_w       = (const float*)d_in[9];
  const float* kernel_att   = (const float*)d_in[10];
  const float* kernel_maxatt= (const float*)d_in[11];
  float* out = (float*)d_out;

  // workspace layout (floats): h_rt | a_lt | a_rt | att | max_att | mp_part
  float* ws      = (float*)d_ws;
  float* ws_h    = ws;                                   // NB*DD
  float* ws_alt  = ws_h    + (size_t)NB * DD;            // NB*LL*AA
  float* ws_art  = ws_alt  + (size_t)NB * LL * AA;       // NB*LR*AA
  float* ws_att  = ws_art  + (size_t)NB * LR * AA;       // NB*LL*DD
  float* ws_max  = ws_att  + (size_t)NB * LL * DD;       // NB*LL*DD
  float* ws_mp   = ws_max  + (size_t)NB * LL * DD;       // 4*NB*LL*MM

  hcat_kernel<<<(NB * DD + 255) / 256, 256, 0, stream>>>(h_rt_fw, h_rt_bw, ws_h);

  dim3 gproj(LL / 16, NB, 2);
  attn_proj_kernel<<<gproj, 256, 0, stream>>>(reps_lt, reps_rt, attn_w1, diag_w,
                                              ws_alt, ws_art);
  dim3 g16(LL / 16, NB);
  attention_kernel<<<g16, 256, 0, stream>>>(ws_alt, ws_art, reps_rt, ws_att);

  dim3 gmax(LL, NB);
  maxatt_kernel<<<gmax, LR, 0, stream>>>(reps_lt, reps_rt, ws_max);

  // max-pooling match: 4 r-quarters in parallel, then combine
  dim3 gmp(LL / 16, NB, 4);
  mpool_part_kernel<<<gmp, 256, 0, stream>>>(reps_lt, reps_rt, kernel_mpool, ws_mp);
  mpool_combine_kernel<<<(NB * LL * MM + 255) / 256, 256, 0, stream>>>(ws_mp, out);

  // full match (h_rt broadcast over l), out channels [0, 65)
  match_kernel<<<g16, 128, 0, stream>>>(reps_lt, ws_h, 0, kernel_full, out, 0);
  // attentive match, out channels [129, 194)
  match_kernel<<<g16, 128, 0, stream>>>(reps_lt, ws_att, DD, kernel_att, out, 129);
  // max-attentive match, out channels [194, 259)
  match_kernel<<<g16, 128, 0, stream>>>(reps_lt, ws_max, DD, kernel_maxatt, out, 194);
}